// WassersteinLoss_420906795339
// MI455X (gfx1250) — compile-verified
//
#include <hip/hip_runtime.h>
#include <hip/hip_bf16.h>

typedef __attribute__((ext_vector_type(2))) float v2f;
typedef __attribute__((ext_vector_type(8))) float v8f;

#define BATCH      8388608      // 2^23 rows
#define NQUADS     (BATCH / 4)  // 4 rows per quad (12 floats = 3 x float4)
#define MAIN_GRID  2048
#define MAIN_BLOCK 256
#define NPART      MAIN_GRID    // one partial per block; 2048 = 32 WMMAs x 64

// fast reciprocal: v_rcp_f32 (~1 ulp). rcp(0) = +inf, matching 1/0 in the ref.
__device__ __forceinline__ float frcp(float x) {
    return __builtin_amdgcn_rcpf(x);
}

// Stable compare-exchange for descending order: swap iff strictly a < b.
// Adjacent-only network with strict compare == stable sort.
#define CSWAP(ka, kb, wa, wb)                    \
    {                                            \
        bool m = (ka) < (kb);                    \
        float tk = ka, tw = wa;                  \
        ka = m ? (kb) : (ka);                    \
        kb = m ? tk : (kb);                      \
        wa = m ? (wb) : (wa);                    \
        wb = m ? tw : (wb);                      \
    }

// One row of the Wasserstein-1 distance (see reference):
//   supports k = [p0, p1, t0, t1], weights w = [p2, 1-p2, -t2, -(1-t2)]
//   stable-descending sort pairs by k, then sum_i |cumsum(w)[i]| * |1/k[i+1]-1/k[i]|
__device__ __forceinline__ float w1_row(float p0, float p1, float p2,
                                        float t0, float t1, float t2) {
    float k0 = p0, k1 = p1, k2 = t0, k3 = t1;
    float w0 = p2, w1 = 1.0f - p2, w2 = -t2, w3 = t2 - 1.0f;
    // bubble network (stable): (0,1)(1,2)(2,3)(0,1)(1,2)(0,1)
    CSWAP(k0, k1, w0, w1);
    CSWAP(k1, k2, w1, w2);
    CSWAP(k2, k3, w2, w3);
    CSWAP(k0, k1, w0, w1);
    CSWAP(k1, k2, w1, w2);
    CSWAP(k0, k1, w0, w1);

    float r0 = frcp(k0), r1 = frcp(k1), r2 = frcp(k2), r3 = frcp(k3);
    float c0 = w0;
    float c1 = c0 + w1;
    float c2 = c1 + w2;
    return fabsf(c0) * fabsf(r1 - r0) +
           fabsf(c1) * fabsf(r2 - r1) +
           fabsf(c2) * fabsf(r3 - r2);
}

// ---------------- main streaming pass: HBM-bound, b128 loads ----------------
__global__ void __launch_bounds__(MAIN_BLOCK)
wloss_main(const float* __restrict__ pred, const float* __restrict__ tgt,
           float* __restrict__ partial) {
    const float4* p4 = reinterpret_cast<const float4*>(pred);
    const float4* t4 = reinterpret_cast<const float4*>(tgt);

    int tid = blockIdx.x * blockDim.x + threadIdx.x;
    int stride = gridDim.x * blockDim.x;

    float acc = 0.0f;
    for (int q = tid; q < NQUADS; q += stride) {
        // 4 rows = 12 floats from each input: three 16B loads each, 16B aligned
        float4 pa = p4[3 * q + 0];
        float4 pb = p4[3 * q + 1];
        float4 pc = p4[3 * q + 2];
        float4 ta = t4[3 * q + 0];
        float4 tb = t4[3 * q + 1];
        float4 tc = t4[3 * q + 2];

        acc += w1_row(pa.x, pa.y, pa.z, ta.x, ta.y, ta.z);
        acc += w1_row(pa.w, pb.x, pb.y, ta.w, tb.x, tb.y);
        acc += w1_row(pb.z, pb.w, pc.x, tb.z, tb.w, tc.x);
        acc += w1_row(pc.y, pc.z, pc.w, tc.y, tc.z, tc.w);
    }

    // deterministic block reduction (wave32 shfl + LDS)
    __shared__ float sred[MAIN_BLOCK / 32];
    float v = acc;
    #pragma unroll
    for (int off = 16; off; off >>= 1) v += __shfl_xor(v, off, 32);

    int wave = threadIdx.x >> 5;
    int lane = threadIdx.x & 31;
    if (lane == 0) sred[wave] = v;
    __syncthreads();
    if (wave == 0) {
        v = (lane < (MAIN_BLOCK / 32)) ? sred[lane] : 0.0f;
        #pragma unroll
        for (int off = 4; off; off >>= 1) v += __shfl_xor(v, off, 32);
        if (lane == 0) partial[blockIdx.x] = v;
    }
}

// ------------- epilogue: single-wave WMMA accumulator over partials ---------
// D = ones(16x4) x B(4x16) + C  =>  every B element is summed into its column
// of C, replicated over rows. One WMMA eats 64 partials (2 VGPRs x 32 lanes);
// 32 chained WMMAs consume all 2048. Row M=0 (lanes 0-15 of c[0]) holds the
// 16 column sums; rows are identical so a full 32-lane reduce of c[0] gives
// 2x the total.
__global__ void __launch_bounds__(32)
wloss_reduce(const float* __restrict__ partial, float* __restrict__ out,
             float inv_batch) {
    int lane = threadIdx.x;  // one wave32, EXEC all ones

    v2f a;
    a.x = 1.0f;
    a.y = 1.0f;
    v8f c = {};

    #pragma unroll 4
    for (int it = 0; it < NPART / 64; ++it) {
        v2f b;
        b.x = partial[it * 64 + lane];
        b.y = partial[it * 64 + 32 + lane];
        c = __builtin_amdgcn_wmma_f32_16x16x4_f32(
            /*neg_a=*/false, a, /*neg_b=*/false, b,
            /*c_mod=*/(short)0, c, /*reuse_a=*/false, /*reuse_b=*/false);
    }

    float v = c[0];
    #pragma unroll
    for (int off = 16; off; off >>= 1) v += __shfl_xor(v, off, 32);

    if (lane == 0) out[0] = v * 0.5f * inv_batch;  // rows M=0 and M=8 both counted
}

extern "C" void kernel_launch(void* const* d_in, const int* in_sizes, int n_in,
                              void* d_out, int out_size, void* d_ws, size_t ws_size,
                              hipStream_t stream) {
    (void)in_sizes; (void)n_in; (void)out_size; (void)ws_size;
    const float* pred = (const float*)d_in[0];
    const float* tgt  = (const float*)d_in[1];
    float* out = (float*)d_out;
    float* partial = (float*)d_ws;  // NPART floats = 8 KB of scratch

    wloss_main<<<MAIN_GRID, MAIN_BLOCK, 0, stream>>>(pred, tgt, partial);
    wloss_reduce<<<1, 32, 0, stream>>>(partial, out, 1.0f / (float)BATCH);
}